// EmptyGCNModel_3135326126101
// MI455X (gfx1250) — compile-verified
//
#include <hip/hip_runtime.h>
#include <hip/hip_bf16.h>

// ---------------------------------------------------------------------------
// GCN forward for MI455X (gfx1250, wave32).
//
// Cost model (MI455X, 23.3 TB/s HBM, 192 MB L2):
//   - 2x weighted scatter-add over 1.6M edges x 128 feats = 204.8M f32 atomics
//     each pass; feature tables (51 MB) are L2-resident. This dominates.
//   - GEMMs are tiny (<=3.3 GFLOP) -> use full-precision V_WMMA_F32_16X16X4_F32
//     so numerics match the f32 reference; matrix throughput is irrelevant here.
// ---------------------------------------------------------------------------

typedef __attribute__((ext_vector_type(2))) float v2f;
typedef __attribute__((ext_vector_type(8))) float v8f;

#define D_FEAT 128  // input/hidden width (both 128 in this model)

// ---------------------------------------------------------------------------
__global__ void zero_f32(float* __restrict__ p, long long n) {
  long long i = (long long)blockIdx.x * blockDim.x + threadIdx.x;
  long long stride = (long long)gridDim.x * blockDim.x;
  for (; i < n; i += stride) p[i] = 0.0f;
}

// One wave32 per edge: lane l moves features[src, 4l..4l+3] * w into agg[dst].
// Coalesced 512B read per wave; 4 global_atomic_add_f32 per lane.
__global__ void scatter_edges(const float* __restrict__ feat,
                              const int* __restrict__ ei,   // [2,E] (harness: integer -> int*)
                              const float* __restrict__ w,
                              float* __restrict__ agg,
                              float* __restrict__ deg,
                              int E, int doDeg) {
  int wid  = (blockIdx.x * blockDim.x + threadIdx.x) >> 5;  // edge id
  int lane = threadIdx.x & 31;
  if (wid >= E) return;
  long long src = (long long)ei[wid];
  long long dst = (long long)ei[(long long)E + wid];
  float ww = w[wid];
  if (doDeg && lane == 0) atomicAdd(deg + dst, ww);
  const float4 f = *(const float4*)(feat + src * D_FEAT + lane * 4);
  float* out = agg + dst * D_FEAT + lane * 4;
  atomicAdd(out + 0, f.x * ww);
  atomicAdd(out + 1, f.y * ww);
  atomicAdd(out + 2, f.z * ww);
  atomicAdd(out + 3, f.w * ww);
}

// agg[n, :] /= max(deg[n], 1)
__global__ void normalize_rows(float* __restrict__ agg,
                               const float* __restrict__ deg,
                               long long total) {
  long long i = (long long)blockIdx.x * blockDim.x + threadIdx.x;
  if (i >= total) return;
  float d = deg[i >> 7];  // row = i / 128
  agg[i] = agg[i] / fmaxf(d, 1.0f);
}

// ---------------------------------------------------------------------------
// out[m, n] = act( sum_k X[m,k] * W[n,k] + bias[n] ),  via V_WMMA_F32_16X16X4_F32.
//
// Grid: blockIdx.x = M/16 tiles; blockDim.x = 32 * n_tiles (one wave per 16-col
// output tile). The 16xK X tile is staged once per block in LDS (coalesced),
// stride K+1 to avoid bank conflicts, then read in the A-fragment layout:
//   lane l (l<16):  A = { X[m0+l, k+0], X[m0+l, k+1] }
//   lane l (l>=16): A = { X[m0+l-16, k+2], X[m0+l-16, k+3] }
// B fragment mirrors the C/D half-split: v0 holds rows {k+2*hi}, v1 rows {k+2*hi+1},
// with B[k,n] = W[n,k].
// C/D layout (per ISA): VGPR r, lanes 0-15 -> (M = r, N = lane),
//                                lanes 16-31 -> (M = r+8, N = lane-16).
// ---------------------------------------------------------------------------
__global__ void wmma_gemm_xwt(const float* __restrict__ X, int K,
                              const float* __restrict__ W,
                              const float* __restrict__ bias,
                              float* __restrict__ out, int ldout, int Nout,
                              int doRelu) {
  __shared__ float xs[16 * (D_FEAT + 1)];  // max K = 128

  const int tid   = threadIdx.x;
  const int nt    = tid >> 5;        // which 16-col tile this wave owns
  const int lane  = tid & 31;
  const int m0    = blockIdx.x * 16;
  const int n0    = nt * 16;
  const int lm    = lane & 15;
  const int hi    = lane >> 4;       // 0 or 1
  const int ldx   = K + 1;           // LDS row stride (conflict-free-ish)
  const int nthr  = blockDim.x;

  // Stage X[m0..m0+15, 0..K) into LDS, coalesced along K.
  for (int i = tid; i < 16 * K; i += nthr) {
    int r = i / K, c = i - r * K;
    xs[r * ldx + c] = X[(long long)(m0 + r) * K + c];
  }
  __syncthreads();

  // Branchless column guard (keeps EXEC all-ones around the WMMA).
  const int  col   = n0 + lm;
  const int  valid = (col < Nout) ? 1 : 0;
  const int  wrow  = valid ? col : 0;
  const float* Wp  = W + (long long)wrow * K + 2 * hi;
  const float* Xp  = xs + lm * ldx + 2 * hi;

  v8f acc = {0.f, 0.f, 0.f, 0.f, 0.f, 0.f, 0.f, 0.f};
  for (int k = 0; k < K; k += 4) {
    v2f a, b;
    a.x = Xp[k];
    a.y = Xp[k + 1];
    b.x = Wp[k];        // garbage columns never get stored; matmul columns
    b.y = Wp[k + 1];    // are independent, so no masking needed here.
    acc = __builtin_amdgcn_wmma_f32_16x16x4_f32(
        /*neg_a=*/false, a, /*neg_b=*/false, b,
        /*c_mod=*/(short)0, acc, /*reuse_a=*/false, /*reuse_b=*/false);
  }

  const float bv = valid ? bias[col] : 0.0f;
#pragma unroll
  for (int r = 0; r < 8; ++r) {
    int   row = m0 + r + 8 * hi;
    float v   = acc[r] + bv;
    if (doRelu) v = fmaxf(v, 0.0f);
    if (valid) out[(long long)row * ldout + col] = v;
  }
}

// ---------------------------------------------------------------------------
extern "C" void kernel_launch(void* const* d_in, const int* in_sizes, int n_in,
                              void* d_out, int out_size, void* d_ws, size_t ws_size,
                              hipStream_t stream) {
  const float* feat = (const float*)d_in[0];   // [N,128]
  const int*   ei   = (const int*)d_in[1];     // [2,E]
  const float* ew   = (const float*)d_in[2];   // [E]
  const float* W1   = (const float*)d_in[3];   // [128,128]
  const float* b1   = (const float*)d_in[4];   // [128]
  const float* W2   = (const float*)d_in[5];   // [64,128]
  const float* b2   = (const float*)d_in[6];   // [64]
  const float* W3   = (const float*)d_in[7];   // [40,64]
  const float* b3   = (const float*)d_in[8];   // [40]

  const int N = in_sizes[0] / D_FEAT;   // 100000
  const int E = in_sizes[2];            // 1600000

  // Workspace layout (f32): agg [N*128] | deg [N] | hidden [N*128]  (~103 MB)
  float* agg    = (float*)d_ws;
  float* deg    = agg + (size_t)N * D_FEAT;
  float* hidden = deg + N;

  float* emb    = (float*)d_out;               // [N,64]
  float* logits = emb + (size_t)N * 64;        // [N,40]

  const long long aggElems = (long long)N * D_FEAT;
  const int edgeBlocks = (E + 7) / 8;          // 8 waves (8 edges) per 256-thr block
  const int normBlocks = (int)((aggElems + 255) / 256);

  // ---- pass 1: agg1 = segsum(feat[src]*w, dst), deg = segsum(w, dst) ----
  zero_f32<<<2048, 256, 0, stream>>>(agg, aggElems + N);  // agg + deg contiguous
  scatter_edges<<<edgeBlocks, 256, 0, stream>>>(feat, ei, ew, agg, deg, E, 1);
  normalize_rows<<<normBlocks, 256, 0, stream>>>(agg, deg, aggElems);

  // ---- hidden = relu(agg1 @ W1^T + b1), [N,128]x[128,128] ----
  wmma_gemm_xwt<<<N / 16, 32 * 8, 0, stream>>>(agg, 128, W1, b1,
                                               hidden, 128, 128, /*relu=*/1);

  // ---- pass 2: agg2 = segsum(hidden[src]*w, dst) (same deg) ----
  zero_f32<<<2048, 256, 0, stream>>>(agg, aggElems);
  scatter_edges<<<edgeBlocks, 256, 0, stream>>>(hidden, ei, ew, agg, deg, E, 0);
  normalize_rows<<<normBlocks, 256, 0, stream>>>(agg, deg, aggElems);

  // ---- emb = agg2 @ W2^T + b2, [N,128]x[128,64] ----
  wmma_gemm_xwt<<<N / 16, 32 * 4, 0, stream>>>(agg, 128, W2, b2,
                                               emb, 64, 64, /*relu=*/0);

  // ---- logits = emb @ W3^T + b3, [N,64]x[64,40] (3 n-tiles, col-guarded) ----
  wmma_gemm_xwt<<<N / 16, 32 * 3, 0, stream>>>(emb, 64, W3, b3,
                                               logits, 40, 40, /*relu=*/0);
}